// FastEctLayer_1769526526455
// MI455X (gfx1250) — compile-verified
//
#include <hip/hip_runtime.h>
#include <math.h>

typedef __attribute__((ext_vector_type(2))) float v2f;
typedef __attribute__((ext_vector_type(8))) float v8f;

#define RES    64          // RESOLUTION
#define TDIRS  64          // number of directions T
#define CHUNK  2048        // points per workgroup
#define NWAVES 8           // 256 threads / wave32
#define TILES_PER_BLOCK (CHUNK / 16)
#define WINDOW 4           // LDS histogram covers 4 consecutive batch ids (64 KB)

// bin = clip(floor((nh + 1.1)/2.2 * 64), 0, 63)  ==  clip(floor(fma(nh, 64/2.2, 32)), 0, 63)
#define BIN_SCALE 29.090909090909090909f
#define BIN_BIAS  32.0f

// ---------------------------------------------------------------- zero counts
__global__ __launch_bounds__(256) void ect_zero(float* __restrict__ counts, int n) {
  int i = blockIdx.x * blockDim.x + threadIdx.x;
  if (i < n) counts[i] = 0.0f;
}

// ------------------------------------------- project (WMMA) + bin + histogram
template <bool FAST>
__device__ __forceinline__ void ect_tiles(const float* __restrict__ x,
                                          const int*   __restrict__ batch,
                                          float*       __restrict__ counts,
                                          float*       hist,
                                          const v2f*   bf,
                                          int chunk_start, int base_b, int N,
                                          int wave, int lane) {
  const int m   = lane & 15;
  const bool hi = lane >= 16;

  for (int tile = wave; tile < TILES_PER_BLOCK; tile += NWAVES) {
    const int p0 = chunk_start + tile * 16;
    const int pm = p0 + m;

    // A fragment: 16 points x K4 (x,y,z,0); lanes 0-15 hold K0/K1, lanes 16-31 hold K2/K3
    v2f af;
    if (FAST || pm < N) {
      const float* xp = x + 3 * pm;
      if (!hi) { af.x = xp[0]; af.y = xp[1]; }
      else     { af.x = xp[2]; af.y = 0.0f;  }
    } else {
      af.x = 0.0f; af.y = 0.0f;
    }

    // prefetch next tile of x for this wave
    {
      int pn = pm + 16 * NWAVES;
      if (pn < N) __builtin_prefetch(x + 3 * pn, 0, 1);
    }

    // batch id of point (p0 + m), broadcast below via ds_bpermute shuffle
    const int bval = (FAST || pm < N) ? batch[pm] : 0;

    v8f c = {};
    // nh tile: 16 points x 64 directions via four f32 WMMAs (K=4)
    v8f d0 = __builtin_amdgcn_wmma_f32_16x16x4_f32(false, af, false, bf[0], (short)0, c, false, false);
    v8f d1 = __builtin_amdgcn_wmma_f32_16x16x4_f32(false, af, false, bf[1], (short)0, c, false, false);
    v8f d2 = __builtin_amdgcn_wmma_f32_16x16x4_f32(false, af, false, bf[2], (short)0, c, false, false);
    v8f d3 = __builtin_amdgcn_wmma_f32_16x16x4_f32(false, af, false, bf[3], (short)0, c, false, false);

#pragma unroll
    for (int j = 0; j < 8; ++j) {
      const int M = j + (hi ? 8 : 0);        // row of D held in VGPR j of this half-wave
      const int p = p0 + M;
      const int bb = __shfl(bval, M, 32);    // batch id of point p
      const int rel = bb - base_b;           // in [0, WINDOW) when FAST
#pragma unroll
      for (int i = 0; i < 4; ++i) {
        float nh = (i == 0) ? d0[j] : (i == 1) ? d1[j] : (i == 2) ? d2[j] : d3[j];
        int bin = (int)floorf(fmaf(nh, BIN_SCALE, BIN_BIAS));
        bin = bin < 0 ? 0 : (bin > 63 ? 63 : bin);   // v_med3_i32
        const int t = i * 16 + m;                    // D column = direction index
        if (FAST) {
          atomicAdd(&hist[(rel * RES + bin) * TDIRS + t], 1.0f);       // ds_add_f32
        } else if (p < N) {
          if ((unsigned)rel < WINDOW) {
            atomicAdd(&hist[(rel * RES + bin) * TDIRS + t], 1.0f);     // ds_add_f32
          } else {
            atomicAdd(&counts[(bb * RES + bin) * TDIRS + t], 1.0f);    // rare fallback
          }
        }
      }
    }
  }
}

__global__ __launch_bounds__(256) void ect_accum(const float* __restrict__ x,
                                                 const float* __restrict__ v,
                                                 const int*   __restrict__ batch,
                                                 float*       __restrict__ counts,
                                                 int N) {
  __shared__ float hist[WINDOW * RES * TDIRS];   // 64 KB

  const int tid  = threadIdx.x;
  const int lane = tid & 31;
  const int wave = tid >> 5;
  const int m    = lane & 15;
  const bool hi  = lane >= 16;
  const int chunk_start = blockIdx.x * CHUNK;

  // zero the LDS window histogram
  for (int i = tid; i < WINDOW * RES * TDIRS; i += 256) hist[i] = 0.0f;

  // B fragments for v [3,64] -> four 4x16 column tiles (K padded to 4 with 0).
  // Layout mirrors A 16x4: VGPR0 = rows K0(lanes0-15)/K2(lanes16-31), VGPR1 = K1/K3.
  v2f bf[4];
#pragma unroll
  for (int i = 0; i < 4; ++i) {
    int t = i * 16 + m;
    if (!hi) { bf[i].x = v[t];        bf[i].y = v[64 + t]; }
    else     { bf[i].x = v[128 + t];  bf[i].y = 0.0f;      }
  }

  // batch is sorted: window covers [base_b, base_b + WINDOW)
  const int first_idx = chunk_start < N ? chunk_start : (N - 1);
  const int last_idx  = (chunk_start + CHUNK <= N) ? (chunk_start + CHUNK - 1) : (N - 1);
  const int base_b = batch[first_idx];
  const int last_b = batch[last_idx];

  // block-uniform fast path: full chunk in-bounds AND whole batch range inside window
  const bool fast = (chunk_start + CHUNK <= N) && ((unsigned)(last_b - base_b) < WINDOW);

  __syncthreads();

  if (fast) {
    ect_tiles<true >(x, batch, counts, hist, bf, chunk_start, base_b, N, wave, lane);
  } else {
    ect_tiles<false>(x, batch, counts, hist, bf, chunk_start, base_b, N, wave, lane);
  }

  __syncthreads();

  // flush LDS window to global counts, skipping zeros
  for (int i = tid; i < WINDOW * RES * TDIRS; i += 256) {
    float cv = hist[i];
    if (cv != 0.0f) {
      atomicAdd(&counts[base_b * (RES * TDIRS) + i], cv);
    }
  }
}

// -------------------------------------------------- in-place cumsum over bins
__global__ __launch_bounds__(64) void ect_cumsum(float* __restrict__ counts) {
  const int b = blockIdx.x;
  const int t = threadIdx.x;
  float* base = counts + b * RES * TDIRS + t;
  float run = 0.0f;
#pragma unroll 8
  for (int r = 0; r < RES; ++r) {
    run += base[r * TDIRS];
    base[r * TDIRS] = run;
  }
}

extern "C" void kernel_launch(void* const* d_in, const int* in_sizes, int n_in,
                              void* d_out, int out_size, void* d_ws, size_t ws_size,
                              hipStream_t stream) {
  const float* x     = (const float*)d_in[0];   // [N, 3]
  const float* v     = (const float*)d_in[1];   // [3, 64]
  const int*   batch = (const int*)d_in[2];     // [N], sorted
  float* counts      = (float*)d_out;           // [B, RES, TDIRS] = out_size floats

  const int N = in_sizes[0] / 3;
  const int B = out_size / (RES * TDIRS);       // num_batches without touching device scalar

  ect_zero<<<(out_size + 255) / 256, 256, 0, stream>>>(counts, out_size);

  const int nblocks = (N + CHUNK - 1) / CHUNK;
  ect_accum<<<nblocks, 256, 0, stream>>>(x, v, batch, counts, N);

  ect_cumsum<<<B, 64, 0, stream>>>(counts);
}